// FeatureOnlyGate_12635793784886
// MI455X (gfx1250) — compile-verified
//
#include <hip/hip_runtime.h>
#include <math.h>

typedef __attribute__((ext_vector_type(2))) float v2f;
typedef __attribute__((ext_vector_type(8))) float v8f;

#define EMB 64
#define NEXP 6
#define TILES 4                    // 16-row WMMA tiles per wave (must be even)
#define WAVES 8                    // waves per block (wave32 -> 256 threads)
#define ROWS_PER_BLOCK (WAVES * TILES * 16)   // 512

__global__ __launch_bounds__(WAVES * 32)
void gate_topk_kernel(const float* __restrict__ h,
                      const float* __restrict__ W,
                      const float* __restrict__ b,
                      float* __restrict__ out,
                      int n) {
    const int lane = threadIdx.x & 31;
    const int wave = threadIdx.x >> 5;
    const int m    = lane & 15;    // row-in-tile (B) / expert row (A)
    const int half = lane >> 4;    // which K-pair within the K=4 step

    // ---- Preload A = W with rows 6..15 clamped to row 5 (no zero-padding
    // needed: C rows M>=6 are never read, so duplicated-expert garbage there
    // is harmless and finite). A layout (32-bit 16x4): lanes 0-15 hold
    // {K=k0,k0+1}, lanes 16-31 hold {k0+2,k0+3}.
    const int wrow = (m < NEXP) ? m : (NEXP - 1);
    const float* wp = W + wrow * EMB + 2 * half;
    v2f a[16];
#pragma unroll
    for (int s = 0; s < 16; ++s) a[s] = *(const v2f*)(wp + 4 * s);

    const float b0 = b[0], b1 = b[1], b2 = b[2], b3 = b[3], b4 = b[4], b5 = b[5];

    const unsigned rowBase = (unsigned)blockIdx.x * ROWS_PER_BLOCK
                           + (unsigned)wave * (TILES * 16);

    // 32-bit element offsets (uniform base + voffset addressing); clamped so
    // loads stay in range (grid fits exactly for this N -> clamp is a no-op).
    unsigned off[TILES];
#pragma unroll
    for (int t = 0; t < TILES; ++t) {
        unsigned r = rowBase + t * 16 + m;
        if (r >= (unsigned)n) r = (unsigned)n - 1;
        off[t] = r * EMB + 2 * half;
    }

    v8f c[TILES];
#pragma unroll
    for (int t = 0; t < TILES; ++t) c[t] = (v8f){0, 0, 0, 0, 0, 0, 0, 0};

    // ---- K loop: 16 steps of V_WMMA_F32_16X16X4_F32 per tile.
    // B layout (32-bit 4x16): lane n%16 holds column n; the two VGPRs hold the
    // two K's of this half.
#pragma unroll
    for (int s = 0; s < 16; ++s) {
        v2f bf[TILES];
#pragma unroll
        for (int t = 0; t < TILES; ++t)
            bf[t] = *(const v2f*)(h + off[t] + 4 * s);
#pragma unroll
        for (int t = 0; t < TILES; ++t)
            c[t] = __builtin_amdgcn_wmma_f32_16x16x4_f32(
                false, a[s], false, bf[t], (short)0, c[t], false, false);
    }

    // ---- Epilogue over tile PAIRS with all 32 lanes active.
    // After WMMA, lane r (0-15) holds experts 0..5 of row (tileRow + r) in
    // c[t][0..5]. For pair p = (t0=2p, t1=2p+1): lanes 0-15 keep t0's logits,
    // lanes 16-31 fetch t1's logits from lane (lane-16) via ds_bpermute.
    // Then row = rowBase + 32p + lane uniformly across the wave.
    const int pidx = (lane & 15) * 4;   // byte lane index for ds_bpermute
#pragma unroll
    for (int p = 0; p < TILES / 2; ++p) {
        const int t0 = 2 * p, t1 = 2 * p + 1;
        float g[NEXP];
#pragma unroll
        for (int e = 0; e < NEXP; ++e) {
            const int pv = __builtin_amdgcn_ds_bpermute(pidx, __float_as_int(c[t1][e]));
            g[e] = (half == 0) ? c[t0][e] : __int_as_float(pv);
        }
        const float g0 = g[0] + b0, g1 = g[1] + b1, g2 = g[2] + b2;
        const float g3 = g[3] + b3, g4 = g[4] + b4, g5 = g[5] + b5;

        // argmax (first index wins ties, matching lax.top_k)
        float ga = g0; int ia = 0;
        if (g1 > ga) { ga = g1; ia = 1; }
        if (g2 > ga) { ga = g2; ia = 2; }
        if (g3 > ga) { ga = g3; ia = 3; }
        if (g4 > ga) { ga = g4; ia = 4; }
        if (g5 > ga) { ga = g5; ia = 5; }
        // second max over e != ia
        float gb = -INFINITY; int ib = 0;
        if (ia != 0 && g0 > gb) { gb = g0; ib = 0; }
        if (ia != 1 && g1 > gb) { gb = g1; ib = 1; }
        if (ia != 2 && g2 > gb) { gb = g2; ib = 2; }
        if (ia != 3 && g3 > gb) { gb = g3; ib = 3; }
        if (ia != 4 && g4 > gb) { gb = g4; ib = 4; }
        if (ia != 5 && g5 > gb) { gb = g5; ib = 5; }

        // softmax + top-2 mask + renorm == softmax over just the top-2 logits
        const float texp = __expf(gb - ga);                 // <= 1
        const float d    = __fdividef(1.0f, 1.0f + texp);
        const float pa   = d;                               // top-1 weight
        const float pb   = texp * d;                        // top-2 weight

        const float o0 = (ia == 0) ? pa : (ib == 0) ? pb : 0.0f;
        const float o1 = (ia == 1) ? pa : (ib == 1) ? pb : 0.0f;
        const float o2 = (ia == 2) ? pa : (ib == 2) ? pb : 0.0f;
        const float o3 = (ia == 3) ? pa : (ib == 3) ? pb : 0.0f;
        const float o4 = (ia == 4) ? pa : (ib == 4) ? pb : 0.0f;
        const float o5 = (ia == 5) ? pa : (ib == 5) ? pb : 0.0f;

        const unsigned row = rowBase + (unsigned)(p * 32) + (unsigned)lane;
        if (row < (unsigned)n) {
            float* op = out + (size_t)row * NEXP;           // 24B rows -> 8B aligned
            *(v2f*)(op + 0) = (v2f){o0, o1};
            *(v2f*)(op + 2) = (v2f){o2, o3};
            *(v2f*)(op + 4) = (v2f){o4, o5};
        }
    }
}

extern "C" void kernel_launch(void* const* d_in, const int* in_sizes, int n_in,
                              void* d_out, int out_size, void* d_ws, size_t ws_size,
                              hipStream_t stream) {
    const float* h = (const float*)d_in[0];   // [N, 64]
    const float* W = (const float*)d_in[1];   // [6, 64]
    const float* b = (const float*)d_in[2];   // [6]
    float* out = (float*)d_out;               // [N, 6]

    const int n = in_sizes[0] / EMB;          // number of rows
    const int grid = (n + ROWS_PER_BLOCK - 1) / ROWS_PER_BLOCK;

    gate_topk_kernel<<<grid, WAVES * 32, 0, stream>>>(h, W, b, out, n);
}